// GNN_node_58488864637367
// MI455X (gfx1250) — compile-verified
//
#include <hip/hip_runtime.h>

typedef __attribute__((ext_vector_type(2))) float v2f;
typedef __attribute__((ext_vector_type(8))) float v8f;

#define ND 128          // channels
#define NNODES 100000   // nodes
#define NEDGES 1600000  // edges
#define NLAYERS 3
#define BN_EPS 1e-5f
#define APAD 132        // LDS row stride (conflict-free: 4*lm+kB covers 64 banks)
#define MROWS 32        // rows per block (2 WMMA M-tiles; B fragments reused 2x)

// ---------------------------------------------------------------- copy z = h
__global__ void copy_f4(const float4* __restrict__ src, float4* __restrict__ dst, int n4) {
    int i = blockIdx.x * blockDim.x + threadIdx.x;
    if (i < n4) dst[i] = src[i];
}

// ------------------------------------------------- zero BN stats accumulator
__global__ void zero_stats(float* __restrict__ stats) {
    int i = threadIdx.x;              // one block of 256 covers 2*ND
    if (i < 2 * ND) stats[i] = 0.0f;
}

// ---------------------------------------- scatter: z[dst] += h[src] per edge
// one wave32 per edge; lane covers 4 channels (32 lanes * float4 = 128 ch)
__global__ void scatter_add(const float* __restrict__ h,
                            const int* __restrict__ srcIdx,
                            const int* __restrict__ dstIdx,
                            float* __restrict__ z, int E) {
    int gid  = blockIdx.x * blockDim.x + threadIdx.x;
    int e    = gid >> 5;
    if (e >= E) return;
    int lane = gid & 31;
    int s = srcIdx[e];
    int d = dstIdx[e];
    const float4* hp = (const float4*)(h + (size_t)s * ND);
    float4 v = hp[lane];
    float* zp = z + (size_t)d * ND + lane * 4;
    atomicAdd(zp + 0, v.x);
    atomicAdd(zp + 1, v.y);
    atomicAdd(zp + 2, v.z);
    atomicAdd(zp + 3, v.w);
}

// ------------------------------------------------- WMMA GEMM: Out = A@W + b
// A:[N,128] row-major, W:[128,128] row-major.
// Block = 256 threads = 8 waves; block owns a 32-row stripe (two 16-row A
// tiles staged in LDS, BN+ReLU fused into staging when FUSE); wave owns a
// 16-column tile with TWO accumulators so each B fragment feeds 2 WMMAs.
// Epilogue fuses per-column sum / sum-of-squares atomics for the next BN.
template <bool FUSE>
__global__ void gemm_wmma(const float* __restrict__ A,
                          const float* __restrict__ W,
                          const float* __restrict__ bias,
                          const float* __restrict__ coeffs,  // [2*ND] scale|shift
                          float* __restrict__ Out,
                          float* __restrict__ stats) {       // [2*ND] sum|sumsq
    __shared__ float As[MROWS * APAD];

    int tid     = threadIdx.x;
    int wave    = tid >> 5;              // 0..7 : column tile
    int lane    = tid & 31;
    int half    = lane >> 4;             // 0 or 1 (K split per ISA layout)
    int lm      = lane & 15;
    int rowBase = blockIdx.x * MROWS;
    int col     = wave * 16 + lm;

    // ---- stage A tile: 32 rows x 128 ch; 256 threads x 16 floats each ----
    {
        int r  = tid >> 3;               // 0..31
        int c0 = (tid & 7) * 16;         // 0,16,...,112
        const float* srcp = A + (size_t)(rowBase + r) * ND + c0;
        float4 v[4];
#pragma unroll
        for (int j = 0; j < 4; ++j) v[j] = *(const float4*)(srcp + 4 * j);
        if (FUSE) {                      // BN1 + ReLU applied once, here
#pragma unroll
            for (int j = 0; j < 4; ++j) {
                const float4 sc = *(const float4*)(coeffs + c0 + 4 * j);
                const float4 sh = *(const float4*)(coeffs + ND + c0 + 4 * j);
                v[j].x = fmaxf(v[j].x * sc.x + sh.x, 0.0f);
                v[j].y = fmaxf(v[j].y * sc.y + sh.y, 0.0f);
                v[j].z = fmaxf(v[j].z * sc.z + sh.z, 0.0f);
                v[j].w = fmaxf(v[j].w * sc.w + sh.w, 0.0f);
            }
        }
        float* dstp = As + r * APAD + c0;     // row stride 528B (16B aligned)
#pragma unroll
        for (int j = 0; j < 4; ++j) *(float4*)(dstp + 4 * j) = v[j];
    }

    // hint L2 about this wave's W column slice (global_prefetch_b8)
    __builtin_prefetch(W + (size_t)half * 64 * ND + col, 0, 1);

    __syncthreads();

    v8f c0v = {};
    v8f c1v = {};
#pragma unroll
    for (int k0 = 0; k0 < ND; k0 += 4) {
        int kB = k0 + 2 * half;
        // B fragment: 4x16 f32 -> lane=N, VGPR pair = K rows (reused 2x)
        v2f b;
        b.x = W[(size_t)kB * ND + col];
        b.y = W[(size_t)(kB + 1) * ND + col];
        // A fragments from LDS: lane=M, contiguous K pair (ds_load_b64)
        const float* ap0 = As + lm * APAD + kB;
        const float* ap1 = As + (16 + lm) * APAD + kB;
        v2f a0, a1;
        a0.x = ap0[0]; a0.y = ap0[1];
        a1.x = ap1[0]; a1.y = ap1[1];
        c0v = __builtin_amdgcn_wmma_f32_16x16x4_f32(false, a0, false, b,
                                                    (short)0, c0v, false, false);
        c1v = __builtin_amdgcn_wmma_f32_16x16x4_f32(false, a1, false, b,
                                                    (short)0, c1v, false, false);
    }

    float bv = bias[col];
    float s = 0.0f, ss = 0.0f;
    float* outp0 = Out + (size_t)(rowBase + 8 * half) * ND + col;
    float* outp1 = Out + (size_t)(rowBase + 16 + 8 * half) * ND + col;
#pragma unroll
    for (int r = 0; r < 8; ++r) {       // VGPR r holds M = r + 8*half
        float v0 = c0v[r] + bv;
        float v1 = c1v[r] + bv;
        s  += v0 + v1;
        ss += v0 * v0 + v1 * v1;
        outp0[(size_t)r * ND] = v0;
        outp1[(size_t)r * ND] = v1;
    }
    atomicAdd(&stats[col], s);
    atomicAdd(&stats[ND + col], ss);
}

// ------------------------------------ turn (sum,sumsq) into (scale,shift)
__global__ void bn_coeffs(const float* __restrict__ stats,
                          const float* __restrict__ gamma,
                          const float* __restrict__ beta,
                          float* __restrict__ coeffs, float invN) {
    int ch = threadIdx.x;
    if (ch >= ND) return;
    float mu  = stats[ch] * invN;
    float var = stats[ND + ch] * invN - mu * mu;
    float sc  = gamma[ch] * rsqrtf(var + BN_EPS);
    coeffs[ch]      = sc;
    coeffs[ND + ch] = beta[ch] - mu * sc;
}

// ------------------------------------------- elementwise BN apply (+ReLU)
__global__ void bn_apply(const float* __restrict__ in,
                         const float* __restrict__ coeffs,
                         float* __restrict__ out, int n, int relu) {
    int i = blockIdx.x * blockDim.x + threadIdx.x;
    if (i >= n) return;
    int ch = i & (ND - 1);
    float v = in[i] * coeffs[ch] + coeffs[ND + ch];
    if (relu) v = fmaxf(v, 0.0f);
    out[i] = v;
}

// ---------------------------------------------------------------------------
extern "C" void kernel_launch(void* const* d_in, const int* in_sizes, int n_in,
                              void* d_out, int out_size, void* d_ws, size_t ws_size,
                              hipStream_t stream) {
    const float* x   = (const float*)d_in[0];
    const int*   ei  = (const int*)d_in[1];     // (2,E): src row then dst row
    const float* W1  = (const float*)d_in[4];   // (L,D,D)
    const float* b1  = (const float*)d_in[5];
    const float* g1  = (const float*)d_in[6];
    const float* be1 = (const float*)d_in[7];
    const float* W2  = (const float*)d_in[8];
    const float* b2  = (const float*)d_in[9];
    const float* g2  = (const float*)d_in[10];
    const float* be2 = (const float*)d_in[11];

    const int* srcIdx = ei;
    const int* dstIdx = ei + NEDGES;

    float* bufA   = (float*)d_ws;                       // N*D
    float* bufB   = bufA + (size_t)NNODES * ND;         // N*D
    float* stats  = bufB + (size_t)NNODES * ND;         // 2*D
    float* coeffs = stats + 2 * ND;                     // 2*D

    const float invN = 1.0f / (float)NNODES;
    const int n4       = NNODES * ND / 4;
    const int copyGrid = (n4 + 255) / 256;
    const int scatGrid = (NEDGES * 32 + 255) / 256;
    const int gemmGrid = NNODES / MROWS;                // 3125 row stripes
    const int elemN    = NNODES * ND;
    const int elemGrid = (elemN + 255) / 256;

    const float* h = x;
    for (int l = 0; l < NLAYERS; ++l) {
        float* z = (l & 1) ? bufB : bufA;
        float* t = (l & 1) ? bufA : bufB;

        // z = h + segment_sum(h[src] -> dst)
        copy_f4<<<copyGrid, 256, 0, stream>>>((const float4*)h, (float4*)z, n4);
        scatter_add<<<scatGrid, 256, 0, stream>>>(h, srcIdx, dstIdx, z, NEDGES);

        // t = z @ W1[l] + b1[l]   (+ BN1 stats)
        zero_stats<<<1, 256, 0, stream>>>(stats);
        gemm_wmma<false><<<gemmGrid, 256, 0, stream>>>(
            z, W1 + (size_t)l * ND * ND, b1 + l * ND, nullptr, t, stats);
        bn_coeffs<<<1, ND, 0, stream>>>(stats, g1 + l * ND, be1 + l * ND, coeffs, invN);

        // u = relu(BN1(t)) @ W2[l] + b2[l]   (+ BN2 stats), u overwrites z
        zero_stats<<<1, 256, 0, stream>>>(stats);
        gemm_wmma<true><<<gemmGrid, 256, 0, stream>>>(
            t, W2 + (size_t)l * ND * ND, b2 + l * ND, coeffs, z, stats);
        bn_coeffs<<<1, ND, 0, stream>>>(stats, g2 + l * ND, be2 + l * ND, coeffs, invN);

        // h_next = BN2(u) (+ReLU except last layer)
        float* dst = (l == NLAYERS - 1) ? (float*)d_out : z;
        bn_apply<<<elemGrid, 256, 0, stream>>>(z, coeffs, dst, elemN,
                                               (l < NLAYERS - 1) ? 1 : 0);
        h = z;
    }
}